// Ende_3332894622093
// MI455X (gfx1250) — compile-verified
//
#include <hip/hip_runtime.h>

// MI455X / gfx1250, wave32. One wave = one sample, 8 waves (256 threads) per block.
// Fused forward coupling layers + structured Jacobian. The per-sample 20x20x20
// matmul JJ12 @ J21 runs on the matrix pipe: V_WMMA_F32_16X16X4_F32, 2x2 output
// tiles held in 4 accumulators, single K loop (5 steps of 4) so A/B fragments
// are each built exactly once per k-step.

typedef __attribute__((ext_vector_type(2))) float v2f;
typedef __attribute__((ext_vector_type(8))) float v8f;

#define HB 20
#define DD 40
#define WPB 8  // waves per block

__device__ __forceinline__ float fast_tanhf(float x) {
  x = fminf(fmaxf(x, -15.f), 15.f);
  const float u = __expf(-2.f * x);                  // v_exp_f32 path
  const float r = __builtin_amdgcn_rcpf(1.f + u);    // v_rcp_f32
  return (1.f - u) * r;
}

__global__ __launch_bounds__(256, 1) void ende_fused(
    const float* __restrict__ src,
    const float* __restrict__ w1, const float* __restrict__ w2,
    const float* __restrict__ w3, const float* __restrict__ w4,
    float* __restrict__ out,    // [B][40]
    float* __restrict__ jout)   // [B][40][40]
{
  __shared__ float sw[4 * HB * HB];        // w1..w4, row-major
  __shared__ float sx[WPB][DD];            // per-wave input x
  __shared__ float ss2[WPB][HB];           // per-wave s2
  __shared__ float sva[WPB][6][HB];        // a1,a2,a3,a4,e2,e4

  const int tid = threadIdx.x;
  for (int i = tid; i < HB * HB; i += 256) {
    sw[0 * HB * HB + i] = w1[i];
    sw[1 * HB * HB + i] = w2[i];
    sw[2 * HB * HB + i] = w3[i];
    sw[3 * HB * HB + i] = w4[i];
  }

  const int wave = tid >> 5;
  const int lane = tid & 31;
  const long long s = (long long)blockIdx.x * WPB + wave;

  // Stage this wave's 40-float input.
  sx[wave][lane] = src[s * DD + lane];
  if (lane < DD - 32) sx[wave][32 + lane] = src[s * DD + 32 + lane];
  __syncthreads();

  const float* W1 = sw;
  const float* W2 = sw + 400;
  const float* W3 = sw + 800;
  const float* W4 = sw + 1200;
  const float* xv = sx[wave];

  // ---- Stage 1: f1/f2 layer (lanes 0..19 produce one element each) ----
  if (lane < HB) {
    float d1 = 0.f, d2 = 0.f;
#pragma unroll
    for (int j = 0; j < HB; j++) {
      const float p = xv[j];
      d1 = fmaf(p, W1[lane * HB + j], d1);
      d2 = fmaf(p, W2[lane * HB + j], d2);
    }
    const float f1 = fast_tanhf(d1), f2 = fast_tanhf(d2);
    const float e2v = __expf(f2);
    const float p2e = xv[HB + lane] * e2v;
    const float s2i = p2e + f1;
    ss2[wave][lane] = s2i;
    sva[wave][0][lane] = 1.f - f1 * f1;            // a1
    sva[wave][1][lane] = p2e * (1.f - f2 * f2);    // a2
    sva[wave][4][lane] = e2v;                      // e2
    __builtin_nontemporal_store(s2i, &out[s * DD + HB + lane]);  // t2 = s2
  }
  __syncthreads();

  // ---- Stage 2: f3/f4 layer ----
  if (lane < HB) {
    float d3 = 0.f, d4 = 0.f;
#pragma unroll
    for (int j = 0; j < HB; j++) {
      const float q = ss2[wave][j];
      d3 = fmaf(q, W3[lane * HB + j], d3);
      d4 = fmaf(q, W4[lane * HB + j], d4);
    }
    const float f3 = fast_tanhf(d3), f4 = fast_tanhf(d4);
    const float e4v = __expf(f4);
    const float s1e = xv[lane] * e4v;
    __builtin_nontemporal_store(s1e + f3, &out[s * DD + lane]);  // t1
    sva[wave][2][lane] = 1.f - f3 * f3;            // a3
    sva[wave][3][lane] = s1e * (1.f - f4 * f4);    // a4
    sva[wave][5][lane] = e4v;                      // e4
  }
  __syncthreads();

  // ---- Jacobian ----
  const float* a1 = sva[wave][0];
  const float* a2 = sva[wave][1];
  const float* a3 = sva[wave][2];
  const float* a4 = sva[wave][3];
  const float* e2 = sva[wave][4];
  const float* e4 = sva[wave][5];
  float* J = jout + s * (DD * DD);

  const int half = lane >> 4;      // 0: lanes 0-15, 1: lanes 16-31
  const int l16 = lane & 15;

  // A-matrix (JJ12) row multipliers. mi=0 rows = l16 (always valid);
  // mi=1 rows = 16 + l16, valid only for l16 < 4 (clamped index for safe LDS reads).
  const float rm3_0 = a3[l16], rm4_0 = a4[l16];
  const int ar1 = 16 + (l16 & 3);
  const bool ok4 = (l16 < 4);
  const float rm3_1 = a3[ar1], rm4_1 = a4[ar1];
  const int nc1 = 16 + (l16 & 3);       // ni=1 B columns (clamped)

  // diag(e4) folded into the C initializer of the diagonal tiles.
  const float e4_l = e4[l16];
  const float e4_h = e4[ar1];
  v8f acc00, acc01 = {}, acc10 = {}, acc11 = {};
#pragma unroll
  for (int r = 0; r < 8; r++)
    acc00[r] = (half * 8 + r == l16) ? e4_l : 0.f;          // rows 0..15 diag
#pragma unroll
  for (int r = 0; r < 4; r++)
    acc11[r] = (r == l16) ? e4_h : 0.f;                     // rows 16..19 diag

#pragma unroll
  for (int k = 0; k < 5; k++) {
    const int K = 4 * k + 2 * half;   // frag layout: v0 holds K / K+2, v1 holds K+1 / K+3

    // A fragments: JJ12(row, K..K+1) = a3[row]*W3[row,K..] + a4[row]*W4[row,K..]
    const v2f w3p0 = *(const v2f*)&W3[l16 * HB + K];
    const v2f w4p0 = *(const v2f*)&W4[l16 * HB + K];
    const v2f af0 = w3p0 * rm3_0 + w4p0 * rm4_0;
    const v2f w3p1 = *(const v2f*)&W3[ar1 * HB + K];
    const v2f w4p1 = *(const v2f*)&W4[ar1 * HB + K];
    const v2f af1t = w3p1 * rm3_1 + w4p1 * rm4_1;
    v2f af1;
    af1.x = ok4 ? af1t.x : 0.f;
    af1.y = ok4 ? af1t.y : 0.f;

    // B fragments: J21(K..K+1, col) = a1[kk]*W1[kk,col] + a2[kk]*W2[kk,col]
    const v2f a1p = *(const v2f*)&a1[K];
    const v2f a2p = *(const v2f*)&a2[K];
    v2f bf0;
    bf0.x = a1p.x * W1[K * HB + l16] + a2p.x * W2[K * HB + l16];
    bf0.y = a1p.y * W1[(K + 1) * HB + l16] + a2p.y * W2[(K + 1) * HB + l16];
    const float b10 = a1p.x * W1[K * HB + nc1] + a2p.x * W2[K * HB + nc1];
    const float b11 = a1p.y * W1[(K + 1) * HB + nc1] + a2p.y * W2[(K + 1) * HB + nc1];
    v2f bf1;
    bf1.x = ok4 ? b10 : 0.f;
    bf1.y = ok4 ? b11 : 0.f;

    acc00 = __builtin_amdgcn_wmma_f32_16x16x4_f32(false, af0, false, bf0, (short)0, acc00, false, false);
    acc01 = __builtin_amdgcn_wmma_f32_16x16x4_f32(false, af0, false, bf1, (short)0, acc01, false, false);
    acc10 = __builtin_amdgcn_wmma_f32_16x16x4_f32(false, af1, false, bf0, (short)0, acc10, false, false);
    acc11 = __builtin_amdgcn_wmma_f32_16x16x4_f32(false, af1, false, bf1, (short)0, acc11, false, false);
  }

  // TL stores; predication grouped by r-invariant conditions (3 small guarded blocks).
#pragma unroll
  for (int r = 0; r < 8; r++)          // rows 0..15, cols 0..15: all valid
    __builtin_nontemporal_store(acc00[r], &J[(half * 8 + r) * DD + l16]);
  if (l16 < 4) {                       // rows 0..15, cols 16..19
#pragma unroll
    for (int r = 0; r < 8; r++)
      __builtin_nontemporal_store(acc01[r], &J[(half * 8 + r) * DD + 16 + l16]);
  }
  if (half == 0) {                     // rows 16..19, cols 0..15
#pragma unroll
    for (int r = 0; r < 4; r++)
      __builtin_nontemporal_store(acc10[r], &J[(16 + r) * DD + l16]);
  }
  if (half == 0 && l16 < 4) {          // rows 16..19, cols 16..19
#pragma unroll
    for (int r = 0; r < 4; r++)
      __builtin_nontemporal_store(acc11[r], &J[(16 + r) * DD + 16 + l16]);
  }

  // TR = JJ12 * e2[col], BL = J21, BR = diag(e2): float2-vectorized, b64 stores.
  for (int t = lane; t < HB * HB / 2; t += 32) {
    const int r = t / (HB / 2);
    const int c2 = (t - r * (HB / 2)) * 2;            // even column pair
    const int wi = r * HB + c2;
    const v2f w1p = *(const v2f*)&W1[wi];
    const v2f w2p = *(const v2f*)&W2[wi];
    const v2f w3p = *(const v2f*)&W3[wi];
    const v2f w4p = *(const v2f*)&W4[wi];
    const float A1r = a1[r], A2r = a2[r], A3r = a3[r], A4r = a4[r];
    const v2f j21 = w1p * A1r + w2p * A2r;
    const v2f jj12 = w3p * A3r + w4p * A4r;
    const v2f e2p = *(const v2f*)&e2[c2];
    const float e2r = e2[r];
    v2f br;
    br.x = (c2 == r) ? e2r : 0.f;
    br.y = (c2 + 1 == r) ? e2r : 0.f;
    __builtin_nontemporal_store(jj12 * e2p, (v2f*)&J[r * DD + HB + c2]);
    __builtin_nontemporal_store(j21, (v2f*)&J[(HB + r) * DD + c2]);
    __builtin_nontemporal_store(br, (v2f*)&J[(HB + r) * DD + HB + c2]);
  }
}

extern "C" void kernel_launch(void* const* d_in, const int* in_sizes, int n_in,
                              void* d_out, int out_size, void* d_ws, size_t ws_size,
                              hipStream_t stream) {
  const float* src = (const float*)d_in[0];
  const float* w1 = (const float*)d_in[1];
  const float* w2 = (const float*)d_in[2];
  const float* w3 = (const float*)d_in[3];
  const float* w4 = (const float*)d_in[4];
  const int B = in_sizes[0] / DD;            // 65536
  float* out = (float*)d_out;                // [B][1][40]
  float* jout = out + (long long)B * DD;     // [B][40][40]
  const int blocks = B / WPB;                // one wave per sample
  ende_fused<<<blocks, 256, 0, stream>>>(src, w1, w2, w3, w4, out, jout);
  (void)n_in; (void)out_size; (void)d_ws; (void)ws_size;
}